// CurvatureEstimator_35210141892838
// MI455X (gfx1250) — compile-verified
//
#include <hip/hip_runtime.h>
#include <math.h>

// CDNA5 / gfx1250. Fully fused curvature estimator.
// Conv FLOPs on V_WMMA_F32_16X16X4_F32; band matrices held in registers,
// data operands batch-staged from LDS so WMMAs issue in bursts.

typedef float v2f __attribute__((ext_vector_type(2)));
typedef float v8f __attribute__((ext_vector_type(8)));

#define IMG_H 512
#define IMG_W 512
#define CH 8
#define TILE 16
#define HALO 8
#define IN_ROWS 32          // TILE + 2*HALO
#define IN_M 256            // (TILE + 2*HALO) * CH, contiguous in NHWC
#define HXP 257             // pitch: 257 mod 64 == 1 -> conflict-free h-strided reads
#define NPAIR 28
#define GP 48               // padded tap-table pitch (band idx -15..31 -> 0..46)

__global__ __launch_bounds__(256)
void CurvatureEstimator_35210141892838_main(const float* __restrict__ img,
                                            float* __restrict__ part) {
    __shared__ float sIn[IN_ROWS * IN_M];        // raw halo tile        (32 KB)
    __shared__ float sHx[5 * TILE * HXP];        // H-conv results       (~80 KB)
    __shared__ float sA[256 * CH];               // curvature numerator  (8 KB), pixel-major
    __shared__ float sB[256 * CH];               // curvature denominator(8 KB), pixel-major
    __shared__ float sGP[5 * GP];                // zero-padded deriv taps
    __shared__ float sRed[8 * 56];               // cross-wave reduction scratch

    const int tid   = threadIdx.x;
    const int lane  = tid & 31;
    const int wvu   = __builtin_amdgcn_readfirstlane(tid >> 5);  // wave id, scalarized
    const int lm    = lane & 15;
    const int off   = (lane >> 4) << 1;          // K-offset: lanes<16 -> {0,1}, else {2,3}
    const int hbase = (lane >> 4) << 3;          // C/D rows: lanes<16 -> M=v, else M=v+8

    const int tileId = blockIdx.x;
    const int b  = tileId >> 10;
    const int t  = tileId & 1023;
    const int h0 = (t >> 5) << 4;
    const int w0 = (t & 31) << 4;

    // ---- zero-padded Gaussian-derivative taps: sGP[n][15+idx] = g_n[idx], idx in [0,16] ----
    if (tid < 5 * GP) {
        const int n   = tid / GP;
        const int idx = tid % GP - 15;
        float val = 0.f;
        if (idx >= 0 && idx <= 16) {
            const int xi = idx - 8;
            double x = (double)xi, td = x * 0.5;             // sigma = 2
            double g0 = exp(-x * x / 8.0) / sqrt(4.0 * 3.14159265358979323846);
            double he;
            if (n == 0)      he = 1.0;
            else if (n == 1) he = td;
            else if (n == 2) he = td * td - 1.0;
            else if (n == 3) he = td * td * td - 3.0 * td;
            else             he = td * td * td * td - 6.0 * td * td + 3.0;
            double sg = (n & 1) ? -1.0 : 1.0;
            val = (float)((n == 0) ? g0 : sg * he * g0 / (double)(1 << n));
        }
        sGP[tid] = val;
    }

    // ---- load 32 x 256 halo tile (zero-padded at image borders) ----
    const size_t bbase = (size_t)b * (IMG_H * IMG_W * CH);
    for (int q = tid; q < 2048; q += 256) {                  // 2048 float4 = 32 KB
        int r  = q >> 6;
        int m4 = (q & 63) << 2;                              // step 4, never crosses a w
        int gh = h0 - HALO + r;
        int gw = w0 - HALO + (m4 >> 3);
        float4 v = make_float4(0.f, 0.f, 0.f, 0.f);
        if (gh >= 0 && gh < IMG_H && gw >= 0 && gw < IMG_W)
            v = *(const float4*)&img[bbase + ((size_t)gh * IMG_W + gw) * CH + (m4 & 7)];
        *(float4*)&sIn[r * IN_M + m4] = v;
    }
    __syncthreads();

    // ---- band matrices in registers: band[n][j] = taps for K-chunk j of order n.
    //      Same lane layout serves pass-1 A operand and pass-2 B operand.
    v2f band[5][8];
    {
        const int pbase = off - lm + 15;
        #pragma unroll
        for (int k = 0; k < 5; ++k)
            #pragma unroll
            for (int j = 0; j < 8; ++j) {
                const int p = k * GP + 4 * j + pbase;
                band[k][j].x = sGP[p];
                band[k][j].y = sGP[p + 1];
            }
    }

    // ---- pass 1: H-conv.  Out(16h x 16m) = Band(16x32) * In(32 x 16m).
    //      Data operand is shared by all 5 orders -> load once, 2 m-tiles per wave,
    //      two independent WMMA chains interleaved.
    {
        const int m0 = wvu << 4;
        const int m1 = (wvu + 8) << 4;
        v2f d0[8], d1[8];
        #pragma unroll
        for (int j = 0; j < 8; ++j) {
            const int r0 = 4 * j + off;
            d0[j].x = sIn[r0 * IN_M + m0 + lm];
            d0[j].y = sIn[(r0 + 1) * IN_M + m0 + lm];
            d1[j].x = sIn[r0 * IN_M + m1 + lm];
            d1[j].y = sIn[(r0 + 1) * IN_M + m1 + lm];
        }
        #pragma unroll
        for (int k = 0; k < 5; ++k) {
            v8f acc0 = {}, acc1 = {};
            #pragma unroll
            for (int j = 0; j < 8; ++j) {
                acc0 = __builtin_amdgcn_wmma_f32_16x16x4_f32(false, band[k][j], false, d0[j],
                                                             (short)0, acc0, false, false);
                acc1 = __builtin_amdgcn_wmma_f32_16x16x4_f32(false, band[k][j], false, d1[j],
                                                             (short)0, acc1, false, false);
            }
            float* dst = &sHx[k * (TILE * HXP)];
            #pragma unroll
            for (int v = 0; v < 8; ++v) {
                dst[(hbase + v) * HXP + m0 + lm] = acc0[v];
                dst[(hbase + v) * HXP + m1 + lm] = acc1[v];
            }
        }
    }
    __syncthreads();

    // ---- pass 2: W-conv, wave = channel.  Out(16h x 16w) = Hx(16x32) * Band(32x16).
    //      Grouped by source order so each hx tile is loaded once.
    const int c = wvu;
    v8f F[11];
    {
        // field table: f -> (source x-order, y-order)
        const int SRCK[11] = {1,0, 3,2,1,0, 4,3,2,1,0};
        const int GI  [11] = {0,1, 0,1,2,3, 0,1,2,3,4};
        #pragma unroll
        for (int k = 0; k < 5; ++k) {
            const float* src = &sHx[k * (TILE * HXP)];
            v2f adat[8];
            #pragma unroll
            for (int j = 0; j < 8; ++j) {
                const int r0 = 4 * j + off;
                adat[j].x = src[lm * HXP + r0 * CH + c];
                adat[j].y = src[lm * HXP + (r0 + 1) * CH + c];
            }
            #pragma unroll
            for (int f = 0; f < 11; ++f) {
                if (SRCK[f] != k) continue;                  // folds at compile time
                v8f acc = {};
                #pragma unroll
                for (int j = 0; j < 8; ++j)
                    acc = __builtin_amdgcn_wmma_f32_16x16x4_f32(false, adat[j], false,
                                                                band[GI[f]][j],
                                                                (short)0, acc, false, false);
                F[f] = acc;
            }
        }
    }

    // ---- pointwise curvature numerator/denominator per (pixel, channel) ----
    #pragma unroll
    for (int v = 0; v < 8; ++v) {
        float ux = F[0][v], uy = F[1][v];
        float uxxx = F[2][v], uxxy = F[3][v], uxyy = F[4][v], uyyy = F[5][v];
        float uxxxx = F[6][v], uxxxy = F[7][v], uxxyy = F[8][v];
        float uxyyy = F[9][v], uyyyy = F[10][v];
        float ux2 = ux * ux, uy2 = uy * uy;
        float Aval = (uxxxx + uyyyy) * ux2 * uy2
                   + uxxyy * (ux2 * ux2 + uy2 * uy2 - 4.f * ux2 * uy2)
                   + 2.f * ux * uy * (uy2 - ux2) * (uxxxy - uxyyy);
        float B2 = uxxx * ux * ux2 + uyyy * uy * uy2
                 + 3.f * ux * uy * (uxxy * ux + uxyy * uy);
        float Bden = sqrtf(ux2 + uy2) * B2;
        const int pix = (hbase + v) * 16 + lm;
        sA[pix * CH + c] = Aval;
        sB[pix * CH + c] = Bden;
    }
    __syncthreads();

    // ---- per-pixel: dilation (vectorized), edges, shared mask, curvature pairs ----
    const int ph = tid >> 4, pw = tid & 15;
    const int gh = h0 + ph, gw = w0 + pw;
    float dil[CH];
    #pragma unroll
    for (int cc = 0; cc < CH; ++cc) dil[cc] = -3.402823466e+38f;   // -inf padding semantics
    #pragma unroll
    for (int dr = -1; dr <= 1; ++dr) {
        const int hh = gh + dr;
        if (hh < 0 || hh >= IMG_H) continue;
        #pragma unroll
        for (int dw = -1; dw <= 1; ++dw) {
            const int ww = gw + dw;
            if (ww < 0 || ww >= IMG_W) continue;
            const float* p4 = &sIn[(ph + HALO + dr) * IN_M + (pw + HALO + dw) * CH];
            const float4 lo = *(const float4*)p4;
            const float4 hi = *(const float4*)(p4 + 4);
            dil[0] = fmaxf(dil[0], lo.x); dil[1] = fmaxf(dil[1], lo.y);
            dil[2] = fmaxf(dil[2], lo.z); dil[3] = fmaxf(dil[3], lo.w);
            dil[4] = fmaxf(dil[4], hi.x); dil[5] = fmaxf(dil[5], hi.y);
            dil[6] = fmaxf(dil[6], hi.z); dil[7] = fmaxf(dil[7], hi.w);
        }
    }
    float edges[NPAIR], cpe[NPAIR];
    float emax = -3.402823466e+38f;
    {
        int p = 0;
        #pragma unroll
        for (int i = 0; i < CH; ++i)
            #pragma unroll
            for (int j2 = i + 1; j2 < CH; ++j2) {
                edges[p] = dil[i] * dil[j2];
                emax = fmaxf(emax, edges[p]);
                ++p;
            }
    }
    const bool maskv = fabsf(emax) > 1e-6f;
    float curv[CH];
    {
        const float4 A0 = *(const float4*)&sA[tid * CH];
        const float4 A1 = *(const float4*)&sA[tid * CH + 4];
        const float4 B0 = *(const float4*)&sB[tid * CH];
        const float4 B1 = *(const float4*)&sB[tid * CH + 4];
        const float Av[CH] = {A0.x, A0.y, A0.z, A0.w, A1.x, A1.y, A1.z, A1.w};
        const float Bv[CH] = {B0.x, B0.y, B0.z, B0.w, B1.x, B1.y, B1.z, B1.w};
        #pragma unroll
        for (int cc = 0; cc < CH; ++cc)
            curv[cc] = maskv ? (-Av[cc] / Bv[cc]) : 0.f;
    }
    {
        int p = 0;
        #pragma unroll
        for (int i = 0; i < CH; ++i)
            #pragma unroll
            for (int j2 = i + 1; j2 < CH; ++j2) {
                cpe[p] = edges[p] * 0.5f * (curv[i] - curv[j2]);
                ++p;
            }
    }

    // ---- deterministic reduction: wave32 shuffle tree, then fixed-order cross-wave ----
    #pragma unroll
    for (int q = 0; q < 56; ++q) {
        float v = (q < NPAIR) ? cpe[q] : edges[q - NPAIR];
        #pragma unroll
        for (int s = 16; s > 0; s >>= 1) v += __shfl_xor(v, s, 32);
        if (lane == 0) sRed[wvu * 56 + q] = v;
    }
    __syncthreads();
    if (tid < 56) {
        float s = 0.f;
        #pragma unroll
        for (int wq = 0; wq < 8; ++wq) s += sRed[wq * 56 + tid];
        part[(size_t)tileId * 56 + tid] = s;                 // [0..27]=curv sums, [28..55]=edge sums
    }
}

// Final deterministic 1024-way sum + divide -> out[b*28+p]
__global__ void CurvatureEstimator_35210141892838_reduce(const float* __restrict__ part,
                                                         float* __restrict__ out) {
    int idx = threadIdx.x;
    if (idx >= 112) return;
    int b = idx / NPAIR, p = idx % NPAIR;
    const float* base = part + (size_t)b * 1024 * 56;
    float sc = 0.f, se = 0.f;
    for (int t = 0; t < 1024; ++t) {
        sc += base[t * 56 + p];
        se += base[t * 56 + NPAIR + p];
    }
    out[idx] = sc / se;
}

extern "C" void kernel_launch(void* const* d_in, const int* in_sizes, int n_in,
                              void* d_out, int out_size, void* d_ws, size_t ws_size,
                              hipStream_t stream) {
    (void)in_sizes; (void)n_in; (void)out_size; (void)ws_size;
    const float* img = (const float*)d_in[0];     // [4,512,512,8] fp32 NHWC
    float* part = (float*)d_ws;                   // 4096 tiles * 56 floats = 896 KB
    float* out  = (float*)d_out;                  // [4,28] fp32

    CurvatureEstimator_35210141892838_main<<<4096, 256, 0, stream>>>(img, part);
    CurvatureEstimator_35210141892838_reduce<<<1, 128, 0, stream>>>(part, out);
}